// TransformerLanguageModel_46411416600774
// MI455X (gfx1250) — compile-verified
//
#include <hip/hip_runtime.h>
#include <math.h>

#define VSZ    32000
#define DMODEL 1024
#define NHEAD  16
#define NLAYER 8
#define FFDIM  4096
#define TSEQ   1024
#define BATCH  2
#define HDIM   64
#define LN_EPS 1e-5f

typedef __attribute__((ext_vector_type(2))) float v2f;
typedef __attribute__((ext_vector_type(8))) float v8f;

// ---------------------------------------------------------------- embedding
__global__ void embed_kernel(const int* __restrict__ idx,
                             const float* __restrict__ tok,
                             const float* __restrict__ pos,
                             float* __restrict__ x) {
  int i  = blockIdx.x * blockDim.x + threadIdx.x;   // over B*T*D
  int d  = i % DMODEL;
  int bt = i / DMODEL;
  int t  = bt % TSEQ;
  int tk = idx[bt];
  x[i] = tok[(size_t)tk * DMODEL + d] + pos[(size_t)t * DMODEL + d];
}

// ---------------------------------------------------------------- layernorm
// one 256-thread block per row of D=1024
__global__ void ln_kernel(const float* __restrict__ x,
                          const float* __restrict__ w,
                          const float* __restrict__ b,
                          float* __restrict__ out) {
  int row = blockIdx.x;
  int tid = threadIdx.x;
  const float* xr = x + (size_t)row * DMODEL;
  float vals[4];
  float s = 0.f, s2 = 0.f;
#pragma unroll
  for (int i = 0; i < 4; ++i) {
    float vv = xr[tid + 256 * i];
    vals[i] = vv; s += vv; s2 += vv * vv;
  }
#pragma unroll
  for (int off = 16; off; off >>= 1) {
    s  += __shfl_xor(s,  off, 32);
    s2 += __shfl_xor(s2, off, 32);
  }
  __shared__ float sh[16];
  int wid = tid >> 5, lane = tid & 31;
  if (lane == 0) { sh[wid] = s; sh[wid + 8] = s2; }
  __syncthreads();
  if (tid == 0) {
    float ts = 0.f, ts2 = 0.f;
    for (int i = 0; i < 8; ++i) { ts += sh[i]; ts2 += sh[i + 8]; }
    float m  = ts / (float)DMODEL;
    float vr = ts2 / (float)DMODEL - m * m;
    sh[0] = m; sh[1] = rsqrtf(vr + LN_EPS);
  }
  __syncthreads();
  float m = sh[0], r = sh[1];
  float* orow = out + (size_t)row * DMODEL;
#pragma unroll
  for (int i = 0; i < 4; ++i) {
    int d = tid + 256 * i;
    orow[d] = (vals[i] - m) * r * w[d] + b[d];
  }
}

// ---------------------------------------------------------------- WMMA GEMM
// Out[M,N] = A[M,K] @ W[K,N] + bias[N]  (+ residual in-place) (+ ReLU)
// block = 256 threads = 8 waves in a 2x4 grid; block tile 64x128;
// wave tile 32x32 (4 accumulators); K slabs double-buffered via
// GLOBAL_LOAD_ASYNC_TO_LDS_B128 so slab i+1 copies overlap WMMA on slab i.
template <bool HAS_RES, bool RELU>
__global__ __launch_bounds__(256)
void gemm_kernel(const float* __restrict__ A, const float* __restrict__ W,
                 const float* __restrict__ bias, float* __restrict__ Out,
                 int M, int N, int K) {
  const int BM = 64, BN = 128, KC = 32;
  __shared__ float As[2][BM][KC + 4];    // stride 36: conflict-free, 16B rows
  __shared__ float Bs[2][KC][BN + 8];    // stride 136: disjoint half-wave banks

  int tid  = threadIdx.x;
  int lane = tid & 31;
  int wid  = tid >> 5;
  int wm   = wid >> 2;      // 0..1
  int wn   = wid & 3;       // 0..3
  int half = lane >> 4;
  int l    = lane & 15;
  int m0   = blockIdx.y * BM;
  int n0   = blockIdx.x * BN;

  v8f acc[2][2] = {};

  auto stage = [&](int k0, int buf) {
#pragma unroll
    for (int j = 0; j < 2; ++j) {                  // A: 64x32 = 512 float4
      int f = tid + 256 * j;
      int r = f >> 3, c = (f & 7) << 2;
      const float* g = &A[(size_t)(m0 + r) * K + (k0 + c)];
      unsigned lds = (unsigned)(uintptr_t)&As[buf][r][c];
      asm volatile("global_load_async_to_lds_b128 %0, %1, off"
                   :: "v"(lds), "v"(g) : "memory");
    }
#pragma unroll
    for (int j = 0; j < 4; ++j) {                  // B: 32x128 = 1024 float4
      int f = tid + 256 * j;
      int r = f >> 5, c = (f & 31) << 2;
      const float* g = &W[(size_t)(k0 + r) * N + (n0 + c)];
      unsigned lds = (unsigned)(uintptr_t)&Bs[buf][r][c];
      asm volatile("global_load_async_to_lds_b128 %0, %1, off"
                   :: "v"(lds), "v"(g) : "memory");
    }
  };

  stage(0, 0);
  int nslab = K / KC;
  for (int i = 0; i < nslab; ++i) {
    int buf = i & 1;
    if (i + 1 < nslab) {
      stage((i + 1) * KC, buf ^ 1);
      asm volatile("s_wait_asynccnt 0x6" ::: "memory");
    } else {
      asm volatile("s_wait_asynccnt 0x0" ::: "memory");
    }
    __syncthreads();

#pragma unroll
    for (int kk = 0; kk < KC; kk += 4) {
      v2f a0, a1, b0, b1;
      a0.x = As[buf][wm * 32 + l     ][kk + 2 * half + 0];
      a0.y = As[buf][wm * 32 + l     ][kk + 2 * half + 1];
      a1.x = As[buf][wm * 32 + 16 + l][kk + 2 * half + 0];
      a1.y = As[buf][wm * 32 + 16 + l][kk + 2 * half + 1];
      b0.x = Bs[buf][kk + 2 * half + 0][wn * 32 + l];
      b0.y = Bs[buf][kk + 2 * half + 1][wn * 32 + l];
      b1.x = Bs[buf][kk + 2 * half + 0][wn * 32 + 16 + l];
      b1.y = Bs[buf][kk + 2 * half + 1][wn * 32 + 16 + l];
      acc[0][0] = __builtin_amdgcn_wmma_f32_16x16x4_f32(false, a0, false, b0,
                      (short)0, acc[0][0], false, false);
      acc[0][1] = __builtin_amdgcn_wmma_f32_16x16x4_f32(false, a0, false, b1,
                      (short)0, acc[0][1], false, false);
      acc[1][0] = __builtin_amdgcn_wmma_f32_16x16x4_f32(false, a1, false, b0,
                      (short)0, acc[1][0], false, false);
      acc[1][1] = __builtin_amdgcn_wmma_f32_16x16x4_f32(false, a1, false, b1,
                      (short)0, acc[1][1], false, false);
    }
    __syncthreads();
  }

#pragma unroll
  for (int ns = 0; ns < 2; ++ns) {
    int gn = n0 + wn * 32 + ns * 16 + l;
    float bv = bias[gn];
#pragma unroll
    for (int ms = 0; ms < 2; ++ms) {
#pragma unroll
      for (int r = 0; r < 8; ++r) {
        int gm = m0 + wm * 32 + ms * 16 + 8 * half + r;
        float val = acc[ms][ns][r] + bv;
        if (HAS_RES) val += Out[(size_t)gm * N + gn];
        if (RELU)    val  = fmaxf(val, 0.0f);
        Out[(size_t)gm * N + gn] = val;
      }
    }
  }
}

// ---------------------------------------------------------------- attention
// WMMA flash-attention. One wave owns 16 queries; 8 waves/block cover 128
// queries of one (b,h). K/V 16x64 tiles async-staged to LDS (double buffered,
// shared by all waves). Per key tile: S = Q@K^T (16 wmma), online softmax in
// C-layout (row stats per VGPR, 4 shfl_xor within the owning half-wave),
// P transposed through per-wave LDS scratch, O += P@V (16 wmma).
__global__ __launch_bounds__(256)
void attn_kernel(const float* __restrict__ Q, const float* __restrict__ K,
                 const float* __restrict__ V, float* __restrict__ O) {
  __shared__ float Ks[2][16][68];   // stride 68: 16 rows -> 16 distinct banks
  __shared__ float Vs[2][16][72];   // stride 72: 2-row half offset -> +16 banks
  __shared__ float Ps[8][16][20];   // per-wave P scratch, stride 20

  int tid  = threadIdx.x;
  int lane = tid & 31;
  int w    = tid >> 5;              // wave 0..7
  int half = lane >> 4;
  int l    = lane & 15;

  int qb = blockIdx.x & 7;          // 128-query block within (b,h)
  int h  = (blockIdx.x >> 3) & 15;
  int b  = blockIdx.x >> 7;

  int qt = qb * 8 + w;              // this wave's 16-query tile index
  int q0 = qt * 16;

  // preload Q fragments: lane (half,l) needs Q[q0+l][4c+2*half .. +1]
  size_t baseQ = ((size_t)(b * TSEQ + q0 + l)) * DMODEL + h * HDIM;
  v2f qf[16];
#pragma unroll
  for (int c = 0; c < 16; ++c) {
    qf[c].x = Q[baseQ + 4 * c + 2 * half + 0];
    qf[c].y = Q[baseQ + 4 * c + 2 * half + 1];
  }

  v8f o0 = {}, o1 = {}, o2 = {}, o3 = {};
  float mrow[8], srow[8];
#pragma unroll
  for (int r = 0; r < 8; ++r) { mrow[r] = -INFINITY; srow[r] = 0.f; }

  const float scale = 0.125f;       // 1/sqrt(64)
  int nkt = qb * 8 + 8;             // key tiles needed by the whole block

  // stage K/V tile kb into buffer buf: 2 async b128 per thread (256 float4 each)
  auto stage = [&](int kb, int buf) {
    int r = tid >> 4, c = (tid & 15) << 2;
    size_t rowoff = ((size_t)(b * TSEQ + kb * 16 + r)) * DMODEL + h * HDIM + c;
    const float* gk = &K[rowoff];
    unsigned ldsk = (unsigned)(uintptr_t)&Ks[buf][r][c];
    asm volatile("global_load_async_to_lds_b128 %0, %1, off"
                 :: "v"(ldsk), "v"(gk) : "memory");
    const float* gv = &V[rowoff];
    unsigned ldsv = (unsigned)(uintptr_t)&Vs[buf][r][c];
    asm volatile("global_load_async_to_lds_b128 %0, %1, off"
                 :: "v"(ldsv), "v"(gv) : "memory");
  };

  stage(0, 0);
  for (int kb = 0; kb < nkt; ++kb) {
    int buf = kb & 1;
    if (kb + 1 < nkt) {
      stage(kb + 1, buf ^ 1);
      asm volatile("s_wait_asynccnt 0x2" ::: "memory");
    } else {
      asm volatile("s_wait_asynccnt 0x0" ::: "memory");
    }
    __syncthreads();

    if (kb <= qt) {                 // wave-uniform: EXEC stays all-ones
      // ---- S = Q @ K^T : B[kd][key] = K[key][kd]
      v8f sacc = {};
#pragma unroll
      for (int c = 0; c < 16; ++c) {
        v2f bf;
        bf.x = Ks[buf][l][4 * c + 2 * half + 0];
        bf.y = Ks[buf][l][4 * c + 2 * half + 1];
        sacc = __builtin_amdgcn_wmma_f32_16x16x4_f32(false, qf[c], false, bf,
                   (short)0, sacc, false, false);
      }
      // ---- scale + causal mask (diagonal tile: key N=l > query M=r+8*half)
      bool diag = (kb == qt);
      float sv[8];
#pragma unroll
      for (int r = 0; r < 8; ++r) {
        float s = sacc[r] * scale;
        if (diag && (l > r + 8 * half)) s = -INFINITY;
        sv[r] = s;
      }
      // ---- online softmax (row stats live in the 16-lane half owning the row)
      float corr[8];
#pragma unroll
      for (int r = 0; r < 8; ++r) {
        float rm = sv[r];
#pragma unroll
        for (int off = 1; off < 16; off <<= 1) rm = fmaxf(rm, __shfl_xor(rm, off, 32));
        float mnew = fmaxf(mrow[r], rm);
        corr[r] = __expf(mrow[r] - mnew);
        float p  = __expf(sv[r] - mnew);
        mrow[r] = mnew;
        float rs = p;
#pragma unroll
        for (int off = 1; off < 16; off <<= 1) rs += __shfl_xor(rs, off, 32);
        srow[r] = srow[r] * corr[r] + rs;
        Ps[w][r + 8 * half][l] = p;     // C-layout -> LDS (A-layout transpose)
      }
      asm volatile("" ::: "memory");    // keep P stores before P reloads
#pragma unroll
      for (int r = 0; r < 8; ++r) {
        o0[r] *= corr[r]; o1[r] *= corr[r]; o2[r] *= corr[r]; o3[r] *= corr[r];
      }
      // ---- O += P @ V (LDS in-order per wave: stores above are visible)
#pragma unroll
      for (int kc = 0; kc < 16; kc += 4) {
        v2f pf, vf0, vf1, vf2, vf3;
        pf.x  = Ps[w][l][kc + 2 * half + 0];
        pf.y  = Ps[w][l][kc + 2 * half + 1];
        vf0.x = Vs[buf][kc + 2 * half + 0][l];
        vf0.y = Vs[buf][kc + 2 * half + 1][l];
        vf1.x = Vs[buf][kc + 2 * half + 0][16 + l];
        vf1.y = Vs[buf][kc + 2 * half + 1][16 + l];
        vf2.x = Vs[buf][kc + 2 * half + 0][32 + l];
        vf2.y = Vs[buf][kc + 2 * half + 1][32 + l];
        vf3.x = Vs[buf][kc + 2 * half + 0][48 + l];
        vf3.y = Vs[buf][kc + 2 * half + 1][48 + l];
        o0 = __builtin_amdgcn_wmma_f32_16x16x4_f32(false, pf, false, vf0,
                 (short)0, o0, false, false);
        o1 = __builtin_amdgcn_wmma_f32_16x16x4_f32(false, pf, false, vf1,
                 (short)0, o1, false, false);
        o2 = __builtin_amdgcn_wmma_f32_16x16x4_f32(false, pf, false, vf2,
                 (short)0, o2, false, false);
        o3 = __builtin_amdgcn_wmma_f32_16x16x4_f32(false, pf, false, vf3,
                 (short)0, o3, false, false);
      }
    }
    __syncthreads();                // all reads done before buffer overwrite
  }

  // ---- normalize rows and store O tile
  size_t baseO = ((size_t)(b * TSEQ + q0)) * DMODEL + h * HDIM;
#pragma unroll
  for (int r = 0; r < 8; ++r) {
    float inv = 1.0f / srow[r];
    int gm = r + 8 * half;
    float* orow = O + baseO + (size_t)gm * DMODEL;
    orow[l]      = o0[r] * inv;
    orow[16 + l] = o1[r] * inv;
    orow[32 + l] = o2[r] * inv;
    orow[48 + l] = o3[r] * inv;
  }
}

// ---------------------------------------------------------------- LM head
// logits[b, v] = h_last[b] . head_w[:, v] + head_b[v]; one thread per v
__global__ void head_kernel(const float* __restrict__ Hx,
                            const float* __restrict__ Wh,
                            const float* __restrict__ bh,
                            float* __restrict__ out) {
  int v = blockIdx.x * blockDim.x + threadIdx.x;
  int b = blockIdx.y;
  if (v >= VSZ) return;
  const float* hrow = Hx + ((size_t)(b * TSEQ + TSEQ - 1)) * DMODEL;
  float acc = bh[v];
  for (int d = 0; d < DMODEL; ++d) acc += hrow[d] * Wh[(size_t)d * VSZ + v];
  out[(size_t)b * VSZ + v] = acc;
}

// ---------------------------------------------------------------- launcher
extern "C" void kernel_launch(void* const* d_in, const int* in_sizes, int n_in,
                              void* d_out, int out_size, void* d_ws, size_t ws_size,
                              hipStream_t stream) {
  (void)in_sizes; (void)n_in; (void)out_size; (void)ws_size;
  const int*   idx   = (const int*)  d_in[0];
  const float* tok   = (const float*)d_in[1];
  const float* pos   = (const float*)d_in[2];
  const float* ln1w  = (const float*)d_in[3];
  const float* ln1b  = (const float*)d_in[4];
  const float* qw    = (const float*)d_in[5];
  const float* qb    = (const float*)d_in[6];
  const float* kw    = (const float*)d_in[7];
  const float* kb    = (const float*)d_in[8];
  const float* vw    = (const float*)d_in[9];
  const float* vb    = (const float*)d_in[10];
  const float* ow    = (const float*)d_in[11];
  const float* ob    = (const float*)d_in[12];
  const float* ln2w  = (const float*)d_in[13];
  const float* ln2b  = (const float*)d_in[14];
  const float* f1w   = (const float*)d_in[15];
  const float* f1b   = (const float*)d_in[16];
  const float* f2w   = (const float*)d_in[17];
  const float* f2b   = (const float*)d_in[18];
  const float* lnfw  = (const float*)d_in[19];
  const float* lnfb  = (const float*)d_in[20];
  const float* headw = (const float*)d_in[21];
  const float* headb = (const float*)d_in[22];
  float* out = (float*)d_out;

  const int    M   = BATCH * TSEQ;                 // 2048 rows
  const size_t NTD = (size_t)M * DMODEL;           // 2 Mi floats
  float* x  = (float*)d_ws;
  float* h  = x  + NTD;
  float* q  = h  + NTD;
  float* k  = q  + NTD;
  float* v  = k  + NTD;
  float* ao = v  + NTD;
  float* f1 = ao + NTD;                            // M * FF

  dim3 b256(256);
  embed_kernel<<<(M * DMODEL) / 256, b256, 0, stream>>>(idx, tok, pos, x);

  dim3 gD(DMODEL / 128, M / 64);                   // N=1024 GEMMs: (8, 32)
  dim3 gF(FFDIM  / 128, M / 64);                   // N=4096 GEMM:  (32, 32)

  for (int l = 0; l < NLAYER; ++l) {
    const float* Wq = qw + (size_t)l * DMODEL * DMODEL;
    const float* Wk = kw + (size_t)l * DMODEL * DMODEL;
    const float* Wv = vw + (size_t)l * DMODEL * DMODEL;
    const float* Wo = ow + (size_t)l * DMODEL * DMODEL;
    const float* W1 = f1w + (size_t)l * DMODEL * FFDIM;
    const float* W2 = f2w + (size_t)l * FFDIM * DMODEL;

    ln_kernel<<<M, b256, 0, stream>>>(x, ln1w + l * DMODEL, ln1b + l * DMODEL, h);
    gemm_kernel<false, false><<<gD, b256, 0, stream>>>(h, Wq, qb + l * DMODEL, q, M, DMODEL, DMODEL);
    gemm_kernel<false, false><<<gD, b256, 0, stream>>>(h, Wk, kb + l * DMODEL, k, M, DMODEL, DMODEL);
    gemm_kernel<false, false><<<gD, b256, 0, stream>>>(h, Wv, vb + l * DMODEL, v, M, DMODEL, DMODEL);
    attn_kernel<<<BATCH * NHEAD * 8, b256, 0, stream>>>(q, k, v, ao);
    gemm_kernel<true, false><<<gD, b256, 0, stream>>>(ao, Wo, ob + l * DMODEL, x, M, DMODEL, DMODEL);

    ln_kernel<<<M, b256, 0, stream>>>(x, ln2w + l * DMODEL, ln2b + l * DMODEL, h);
    gemm_kernel<false, true ><<<gF, b256, 0, stream>>>(h, W1, f1b + l * FFDIM, f1, M, FFDIM, DMODEL);
    gemm_kernel<true,  false><<<gD, b256, 0, stream>>>(f1, W2, f2b + l * DMODEL, x, M, DMODEL, FFDIM);
  }

  ln_kernel<<<M, b256, 0, stream>>>(x, lnfw, lnfb, h);
  dim3 gh((VSZ + 255) / 256, BATCH);
  head_kernel<<<gh, b256, 0, stream>>>(h, headw, headb, out);
}